// GNNBlock_83356725280827
// MI455X (gfx1250) — compile-verified
//
#include <hip/hip_runtime.h>
#include <hip/hip_bf16.h>
#include <math.h>

#define DIM 128
#define LN_EPS 1e-5f

typedef __attribute__((ext_vector_type(2))) float v2f;
typedef __attribute__((ext_vector_type(8))) float v8f;

// ---------------- init: dinv = 1 (self-loop weight), reduction scalars = 0 ----
__global__ void init_kernel(float* __restrict__ dinv, float* __restrict__ red, int n) {
    int i = blockIdx.x * blockDim.x + threadIdx.x;
    if (i < n) dinv[i] = 1.0f;
    if (i == 0) { red[0] = 0.0f; red[1] = 0.0f; }
}

// ---------------- degree: deg[col] += ew --------------------------------------
__global__ void degree_kernel(const int* __restrict__ col, const float* __restrict__ ew,
                              float* __restrict__ dinv, int E) {
    int e = blockIdx.x * blockDim.x + threadIdx.x;
    if (e < E) atomicAdd(&dinv[col[e]], ew[e]);
}

// ---------------- dinv = rsqrt(deg)  (deg >= 1 always, self-loop) -------------
__global__ void rsqrt_kernel(float* __restrict__ dinv, int n) {
    int i = blockIdx.x * blockDim.x + threadIdx.x;
    if (i < n) dinv[i] = rsqrtf(dinv[i]);
}

// ---------------- Wt[k][j] = W[j][k] ------------------------------------------
__global__ void transpose_kernel(const float* __restrict__ W, float* __restrict__ Wt) {
    int idx = blockIdx.x * blockDim.x + threadIdx.x;  // 0 .. 16383
    int j = idx >> 7;
    int k = idx & (DIM - 1);
    Wt[k * DIM + j] = W[j * DIM + k];
}

// ---------------- h = x @ W^T via V_WMMA_F32_16X16X4_F32 ----------------------
// One 16x16 output tile per wave; 8 waves/block cover 16 rows x 128 cols so all
// waves in the block reuse the same 16 x-rows through WGP$/L2.
__global__ __launch_bounds__(256) void gemm_wmma_kernel(const float* __restrict__ x,
                                                        const float* __restrict__ Wt,
                                                        float* __restrict__ h, int n) {
    const int lane   = threadIdx.x & 31;
    const int wave   = threadIdx.x >> 5;
    const int lo     = lane & 15;      // M (for A) / N (for B,C,D)
    const int hi     = lane >> 4;      // half-wave selector
    const int rowBase = blockIdx.x * 16;
    const int jBase   = wave * 16;
    if (rowBase >= n) return;

    const float* xrow = x + (size_t)(rowBase + lo) * DIM;
    v8f acc = {};  // C starts at 0

    #pragma unroll 4
    for (int k0 = 0; k0 < DIM; k0 += 4) {
        const int kA = k0 + 2 * hi;
        v2f a, b;
        a.x = xrow[kA];
        a.y = xrow[kA + 1];
        b.x = Wt[(size_t)kA * DIM + jBase + lo];
        b.y = Wt[(size_t)(kA + 1) * DIM + jBase + lo];
        acc = __builtin_amdgcn_wmma_f32_16x16x4_f32(
            /*neg_a=*/false, a, /*neg_b=*/false, b,
            /*c_mod=*/(short)0, acc, /*reuse_a=*/false, /*reuse_b=*/false);
    }

    float* hp = h + (size_t)rowBase * DIM + jBase + lo;
    #pragma unroll
    for (int r = 0; r < 8; ++r) {
        int m = r + 8 * hi;
        hp[(size_t)m * DIM] = acc[r];
    }
}

// ---------------- out = h * dinv^2   (self-loop term, full init of out) -------
__global__ void selfloop_kernel(const float* __restrict__ h, const float* __restrict__ dinv,
                                float* __restrict__ out, long total) {
    long idx = (long)blockIdx.x * blockDim.x + threadIdx.x;
    if (idx < total) {
        float di = dinv[idx >> 7];
        out[idx] = h[idx] * di * di;
    }
}

// ---------------- scatter: one edge per wave, float4 gather + f32 atomics -----
__global__ __launch_bounds__(256) void scatter_kernel(const int* __restrict__ row,
                                                      const int* __restrict__ col,
                                                      const float* __restrict__ ew,
                                                      const float* __restrict__ dinv,
                                                      const float* __restrict__ h,
                                                      float* __restrict__ out, int E) {
    int e    = (int)(((long)blockIdx.x * blockDim.x + threadIdx.x) >> 5);
    int lane = threadIdx.x & 31;
    if (e >= E) return;
    int r = row[e];
    int c = col[e];
    float norm = dinv[r] * ew[e] * dinv[c];
    const float4* hp = (const float4*)(h + (size_t)r * DIM);
    float*        op = out + (size_t)c * DIM + lane * 4;
    float4 v = hp[lane];
    atomicAdd(op + 0, v.x * norm);
    atomicAdd(op + 1, v.y * norm);
    atomicAdd(op + 2, v.z * norm);
    atomicAdd(op + 3, v.w * norm);
}

// ---------------- relu(out + b), accumulate sum & sumsq -----------------------
__global__ __launch_bounds__(256) void relu_reduce_kernel(float* __restrict__ out,
                                                          const float* __restrict__ bias,
                                                          float* __restrict__ red, long total) {
    __shared__ float ssum[256];
    __shared__ float ssq[256];
    long stride = (long)gridDim.x * blockDim.x;
    float s = 0.0f, q = 0.0f;
    for (long idx = (long)blockIdx.x * blockDim.x + threadIdx.x; idx < total; idx += stride) {
        float v = out[idx] + bias[idx & (DIM - 1)];
        v = fmaxf(v, 0.0f);
        out[idx] = v;
        s += v;
        q += v * v;
    }
    int tid = threadIdx.x;
    ssum[tid] = s; ssq[tid] = q;
    __syncthreads();
    for (int off = 128; off > 0; off >>= 1) {
        if (tid < off) { ssum[tid] += ssum[tid + off]; ssq[tid] += ssq[tid + off]; }
        __syncthreads();
    }
    if (tid == 0) {
        atomicAdd(&red[0], ssum[0]);
        atomicAdd(&red[1], ssq[0]);
    }
}

// ---------------- graph LayerNorm apply ---------------------------------------
__global__ void ln_apply_kernel(float* __restrict__ out, const float* __restrict__ lnw,
                                const float* __restrict__ lnb, const float* __restrict__ red,
                                long total) {
    long idx = (long)blockIdx.x * blockDim.x + threadIdx.x;
    if (idx < total) {
        float invM = 1.0f / (float)total;
        float mean = red[0] * invM;
        float var  = red[1] * invM - mean * mean;
        float inv  = rsqrtf(var + LN_EPS);
        int d = idx & (DIM - 1);
        out[idx] = (out[idx] - mean) * inv * lnw[d] + lnb[d];
    }
}

extern "C" void kernel_launch(void* const* d_in, const int* in_sizes, int n_in,
                              void* d_out, int out_size, void* d_ws, size_t ws_size,
                              hipStream_t stream) {
    const float* x   = (const float*)d_in[0];
    const int*   ei  = (const int*)d_in[1];
    const float* ew  = (const float*)d_in[2];
    const float* W   = (const float*)d_in[3];
    const float* b   = (const float*)d_in[4];
    const float* lnw = (const float*)d_in[5];
    const float* lnb = (const float*)d_in[6];

    const int N = in_sizes[0] / DIM;
    const int E = in_sizes[2];
    const int* row = ei;       // edge_index[0]
    const int* col = ei + E;   // edge_index[1]
    const long total = (long)N * DIM;

    // workspace layout (floats): h[N*D] | Wt[D*D] | dinv[N] | red[2]
    float* ws   = (float*)d_ws;
    float* h    = ws;
    float* Wt   = h + (size_t)N * DIM;
    float* dinv = Wt + (size_t)DIM * DIM;
    float* red  = dinv + N;

    float* out = (float*)d_out;

    const int B = 256;
    hipLaunchKernelGGL(init_kernel, dim3((N + B - 1) / B), dim3(B), 0, stream, dinv, red, N);
    hipLaunchKernelGGL(degree_kernel, dim3((E + B - 1) / B), dim3(B), 0, stream, col, ew, dinv, E);
    hipLaunchKernelGGL(rsqrt_kernel, dim3((N + B - 1) / B), dim3(B), 0, stream, dinv, N);
    hipLaunchKernelGGL(transpose_kernel, dim3((DIM * DIM) / B), dim3(B), 0, stream, W, Wt);
    hipLaunchKernelGGL(gemm_wmma_kernel, dim3((N + 15) / 16), dim3(B), 0, stream, x, Wt, h, N);
    hipLaunchKernelGGL(selfloop_kernel, dim3((int)((total + B - 1) / B)), dim3(B), 0, stream,
                       h, dinv, out, total);
    hipLaunchKernelGGL(scatter_kernel, dim3((E + 7) / 8), dim3(B), 0, stream,
                       row, col, ew, dinv, h, out, E);
    hipLaunchKernelGGL(relu_reduce_kernel, dim3(2048), dim3(B), 0, stream, out, b, red, total);
    hipLaunchKernelGGL(ln_apply_kernel, dim3((int)((total + B - 1) / B)), dim3(B), 0, stream,
                       out, lnw, lnb, red, total);
}